// SimGNN_566935683386
// MI455X (gfx1250) — compile-verified
//
#include <hip/hip_runtime.h>
#include <math.h>

#define NN   16384
#define EE   262144
#define DIN  144
#define F1D  128
#define F2D  64
#define F3D  32
#define KT   16
#define BINS 16

typedef float v2f __attribute__((ext_vector_type(2)));
typedef float v8f __attribute__((ext_vector_type(8)));

// monotone float<->int encoding for signed-int atomicMin/Max
__device__ __forceinline__ int enc_f(float v) {
    int s = __float_as_int(v);
    return s >= 0 ? s : (s ^ 0x7FFFFFFF);
}
__device__ __forceinline__ float dec_f(int s) {
    return __int_as_float(s >= 0 ? s : (s ^ 0x7FFFFFFF));
}

__device__ __forceinline__ v8f wmma4(v2f a, v2f b, v8f c) {
    return __builtin_amdgcn_wmma_f32_16x16x4_f32(false, a, false, b,
                                                 (short)0, c, false, false);
}

// ---------------- init small accumulators ----------------
__global__ void init_small_kernel(float* dinv1, float* dinv2, float* m1, float* m2,
                                  float* p1, float* p2, unsigned* hist,
                                  int* gmin, int* gmax) {
    int t = blockIdx.x * 256 + threadIdx.x;
    if (t < NN) { dinv1[t] = 0.f; dinv2[t] = 0.f; }
    if (t < F3D) { m1[t] = 0.f; m2[t] = 0.f; p1[t] = 0.f; p2[t] = 0.f; }
    if (t < BINS) hist[t] = 0u;
    if (t == 0) { *gmin = 0x7FFFFFFF; *gmax = (int)0x80000000; }
}

// ---------------- degree / D^-1/2 ----------------
__global__ void deg_kernel(const int* __restrict__ rows, float* deg) {
    int e = blockIdx.x * 256 + threadIdx.x;
    if (e < EE) atomicAdd(&deg[rows[e]], 1.f);
}
__global__ void dinv_kernel(float* deg) {
    int i = blockIdx.x * 256 + threadIdx.x;
    if (i < NN) deg[i] = rsqrtf(fmaxf(deg[i] + 1.f, 1e-12f)); // +1 self loop
}

// ---------------- h = x @ W via V_WMMA_F32_16X16X4_F32 ----------------
// one 16x16 output tile per wave; K-loop over Din in steps of 4.
// A frag: lanes 0-15 K={0,1}, lanes 16-31 K={2,3} -> x row float2 at [kk/2 + half]
// B frag: b.x = W[(kk+2h)*Dout + J+r], b.y = W[(kk+1+2h)*Dout + J+r]
// C store: h[(I + i + 8*half)*Dout + J + r] = c[i]
__global__ void gemm_wmma_kernel(const float* __restrict__ x, const float* __restrict__ W,
                                 float* __restrict__ h, int Din, int Dout) {
    const int wave = threadIdx.x >> 5;
    const int lane = threadIdx.x & 31;
    const int I = (blockIdx.y * 8 + wave) * 16;
    const int J = blockIdx.x * 16;
    const int half = lane >> 4;
    const int r    = lane & 15;
    const v2f*   xr = (const v2f*)(x + (size_t)(I + r) * Din);
    const float* wc = W + J + r;
    v8f c = {0.f, 0.f, 0.f, 0.f, 0.f, 0.f, 0.f, 0.f};
    for (int kk = 0; kk < Din; kk += 4) {
        v2f a = xr[(kk >> 1) + half];
        int kb = kk + 2 * half;
        v2f b;
        b.x = wc[(size_t)kb * Dout];
        b.y = wc[(size_t)(kb + 1) * Dout];
        c = wmma4(a, b, c);
    }
    float* hp = h + (size_t)(I + 8 * half) * Dout + J + r;
#pragma unroll
    for (int i = 0; i < 8; ++i) hp[(size_t)i * Dout] = c[i];
}

// out[i,:] = h[i,:]*dinv[i]^2 + b   (self-loop term + bias)
__global__ void initout_kernel(const float* __restrict__ h, const float* __restrict__ dinv,
                               const float* __restrict__ b, float* __restrict__ out, int Dout) {
    int tid = blockIdx.x * 256 + threadIdx.x;
    int i = tid / Dout, j = tid % Dout;
    float di = dinv[i];
    out[tid] = h[tid] * di * di + b[j];
}

// scatter: out[row,:] += h[col,:] * dinv[row]*dinv[col]
__global__ void agg_kernel(const int* __restrict__ rows, const int* __restrict__ cols,
                           const float* __restrict__ h, const float* __restrict__ dinv,
                           float* __restrict__ out, int Dout) {
    int chunks = Dout >> 2;
    int tid = blockIdx.x * 256 + threadIdx.x;
    int e = tid / chunks;
    int q = tid % chunks;
    if (e >= EE) return;
    int r = rows[e], c = cols[e];
    float nrm = dinv[r] * dinv[c];
    const float4 hv = ((const float4*)(h + (size_t)c * Dout))[q];
    float* op = out + (size_t)r * Dout + q * 4;
    atomicAdd(op + 0, hv.x * nrm);
    atomicAdd(op + 1, hv.y * nrm);
    atomicAdd(op + 2, hv.z * nrm);
    atomicAdd(op + 3, hv.w * nrm);
}

__global__ void relu_kernel(float* x, int n) {
    int tid = blockIdx.x * 256 + threadIdx.x;
    if (tid < n) x[tid] = fmaxf(x[tid], 0.f);
}

// ---------------- attention pooling ----------------
__global__ void colmean_kernel(const float* __restrict__ af, float* m) {
    int t = threadIdx.x;
    int col = t & (F3D - 1);
    int rg  = t >> 5;                               // 0..7
    float s = 0.f;
    for (int i = blockIdx.x * 8 + rg; i < NN; i += gridDim.x * 8)
        s += af[(size_t)i * F3D + col];
    atomicAdd(&m[col], s);
}
__global__ void tg_kernel(const float* __restrict__ m, const float* __restrict__ attW,
                          float* tg) {
    int j = threadIdx.x;
    if (j < F3D) {
        float s = 0.f;
        for (int f = 0; f < F3D; ++f) s += m[f] * attW[f * F3D + j];
        tg[j] = tanhf(s / (float)NN);               // mean(x@W) = mean(x)@W
    }
}
__global__ void attpool_kernel(const float* __restrict__ af, const float* __restrict__ tg,
                               float* pooled) {
    __shared__ float acc[F3D];
    int t = threadIdx.x;
    if (t < F3D) acc[t] = 0.f;
    __syncthreads();
    int i = blockIdx.x * 256 + t;
    const float* row = af + (size_t)i * F3D;
    float d = 0.f;
#pragma unroll
    for (int j = 0; j < F3D; ++j) d += row[j] * tg[j];
    float gate = 1.f / (1.f + expf(-d));
#pragma unroll
    for (int j = 0; j < F3D; ++j) atomicAdd(&acc[j], row[j] * gate);
    __syncthreads();
    if (t < F3D) atomicAdd(&pooled[t], acc[t]);
}

// ------- fused sim GEMM: 2x2 block of 16x16 tiles per wave (32x32 output) -------
// Halves L2 fragment traffic per output: 4 fragment loads feed 4 WMMAs per K-step.
template <bool HIST>
__global__ void sim_tile_kernel(const float* __restrict__ af1, const float* __restrict__ af2,
                                int* gmin, int* gmax, unsigned* hist) {
    const int wave = threadIdx.x >> 5;
    const int lane = threadIdx.x & 31;
    const int I = blockIdx.y * 32;
    const int J = (blockIdx.x * 8 + wave) * 32;
    const int half = lane >> 4;
    const int r    = lane & 15;
    const v2f* a0 = (const v2f*)(af1 + (size_t)(I + r) * F3D);
    const v2f* a1 = (const v2f*)(af1 + (size_t)(I + 16 + r) * F3D);
    const v2f* b0 = (const v2f*)(af2 + (size_t)(J + r) * F3D);
    const v2f* b1 = (const v2f*)(af2 + (size_t)(J + 16 + r) * F3D);
    v8f c00 = {0.f,0.f,0.f,0.f,0.f,0.f,0.f,0.f};
    v8f c01 = c00, c10 = c00, c11 = c00;
#pragma unroll
    for (int kk = 0; kk < 8; ++kk) {            // K = 32 = 8 x (16x16x4)
        v2f A0 = a0[kk * 2 + half];
        v2f A1 = a1[kk * 2 + half];
        v2f B0 = b0[kk * 2 + half];
        v2f B1 = b1[kk * 2 + half];
        c00 = wmma4(A0, B0, c00);
        c01 = wmma4(A0, B1, c01);
        c10 = wmma4(A1, B0, c10);
        c11 = wmma4(A1, B1, c11);
    }
    if (!HIST) {
        float vmin = c00[0], vmax = c00[0];
#pragma unroll
        for (int i = 0; i < 8; ++i) {
            vmin = fminf(vmin, fminf(fminf(c00[i], c01[i]), fminf(c10[i], c11[i])));
            vmax = fmaxf(vmax, fmaxf(fmaxf(c00[i], c01[i]), fmaxf(c10[i], c11[i])));
        }
        __shared__ float smin[256], smax[256];
        smin[threadIdx.x] = vmin; smax[threadIdx.x] = vmax;
        __syncthreads();
        for (int s = 128; s > 0; s >>= 1) {
            if (threadIdx.x < s) {
                smin[threadIdx.x] = fminf(smin[threadIdx.x], smin[threadIdx.x + s]);
                smax[threadIdx.x] = fmaxf(smax[threadIdx.x], smax[threadIdx.x + s]);
            }
            __syncthreads();
        }
        if (threadIdx.x == 0) {
            atomicMin(gmin, enc_f(smin[0]));
            atomicMax(gmax, enc_f(smax[0]));
        }
    } else {
        __shared__ unsigned lh[BINS];
        if (threadIdx.x < BINS) lh[threadIdx.x] = 0u;
        __syncthreads();
        const float vmin  = dec_f(*gmin);
        const float vmax  = dec_f(*gmax);
        const float scale = (vmax > vmin) ? (float)BINS / (vmax - vmin) : 0.f;
        const v8f* cs[4] = { &c00, &c01, &c10, &c11 };
#pragma unroll
        for (int t = 0; t < 4; ++t) {
#pragma unroll
            for (int i = 0; i < 8; ++i) {
                int b = (int)(((*cs[t])[i] - vmin) * scale);
                b = b < 0 ? 0 : (b > BINS - 1 ? BINS - 1 : b);
                atomicAdd(&lh[b], 1u);
            }
        }
        __syncthreads();
        if (threadIdx.x < BINS) atomicAdd(&hist[threadIdx.x], lh[threadIdx.x]);
    }
}

// ---------------- tensor network + histogram norm + FC head ----------------
__global__ void final_kernel(const float* __restrict__ p1v, const float* __restrict__ p2v,
                             const float* __restrict__ tW, const float* __restrict__ tWb,
                             const float* __restrict__ tb, const float* __restrict__ fcW,
                             const float* __restrict__ fcb, const float* __restrict__ scW,
                             const float* __restrict__ scb, const unsigned* __restrict__ hist,
                             float* out) {
    __shared__ float e1[F3D], e2[F3D];
    __shared__ float vec[KT + BINS];
    __shared__ float fco[16];
    __shared__ float tot;
    int t = threadIdx.x;
    if (t < F3D) { e1[t] = p1v[t]; e2[t] = p2v[t]; }
    if (t == 0) {
        unsigned s = 0;
        for (int i = 0; i < BINS; ++i) s += hist[i];
        tot = (float)s;
    }
    __syncthreads();
    if (t < KT) {
        float s = tb[t];
        for (int i = 0; i < F3D; ++i) {
            float a = e1[i];
            for (int f = 0; f < F3D; ++f)
                s += a * tW[(size_t)(i * F3D + f) * KT + t] * e2[f];
        }
        for (int j = 0; j < 2 * F3D; ++j) {
            float cv = (j < F3D) ? e1[j] : e2[j - F3D];
            s += tWb[t * (2 * F3D) + j] * cv;
        }
        vec[t]      = fmaxf(s, 0.f);
        vec[KT + t] = (float)hist[t] / tot;
    }
    __syncthreads();
    if (t < 16) {
        float f = fcb[t];
        for (int j = 0; j < KT + BINS; ++j) f += fcW[t * (KT + BINS) + j] * vec[j];
        fco[t] = fmaxf(f, 0.f);
    }
    __syncthreads();
    if (t == 0) {
        float o = scb[0];
        for (int i = 0; i < 16; ++i) o += scW[i] * fco[i];
        out[0] = 1.f / (1.f + expf(-o));
    }
}

// ==================================================================
extern "C" void kernel_launch(void* const* d_in, const int* in_sizes, int n_in,
                              void* d_out, int out_size, void* d_ws, size_t ws_size,
                              hipStream_t stream) {
    (void)in_sizes; (void)n_in; (void)out_size; (void)ws_size;
    const float* feat1 = (const float*)d_in[0];
    const float* feat2 = (const float*)d_in[1];
    const int*   ei1   = (const int*)d_in[2];
    const int*   ei2   = (const int*)d_in[3];
    const float* W1 = (const float*)d_in[4];
    const float* b1 = (const float*)d_in[5];
    const float* W2 = (const float*)d_in[6];
    const float* b2 = (const float*)d_in[7];
    const float* W3 = (const float*)d_in[8];
    const float* b3 = (const float*)d_in[9];
    const float* attW = (const float*)d_in[10];
    const float* tW   = (const float*)d_in[11];
    const float* tWb  = (const float*)d_in[12];
    const float* tb   = (const float*)d_in[13];
    const float* fcW  = (const float*)d_in[14];
    const float* fcb  = (const float*)d_in[15];
    const float* scW  = (const float*)d_in[16];
    const float* scb  = (const float*)d_in[17];
    float* out = (float*)d_out;

    // workspace carve-out (~25.4 MB)
    float* ws = (float*)d_ws;
    size_t off = 0;
    auto alloc = [&](size_t n) { float* p = ws + off; off += n; return p; };
    float* A1 = alloc((size_t)NN * F1D);   // graph1 ping (final af1, stride 32)
    float* A2 = alloc((size_t)NN * F1D);   // graph2 ping (final af2)
    float* Bs = alloc((size_t)NN * F1D);   // shared pong (x@W scratch)
    float* dinv1 = alloc(NN);
    float* dinv2 = alloc(NN);
    float* m1 = alloc(F3D); float* m2 = alloc(F3D);
    float* tg1 = alloc(F3D); float* tg2 = alloc(F3D);
    float* p1 = alloc(F3D); float* p2 = alloc(F3D);
    int* gmin = (int*)alloc(1);
    int* gmax = (int*)alloc(1);
    unsigned* hist = (unsigned*)alloc(BINS);

    init_small_kernel<<<NN / 256, 256, 0, stream>>>(dinv1, dinv2, m1, m2, p1, p2,
                                                    hist, gmin, gmax);

    deg_kernel<<<EE / 256, 256, 0, stream>>>(ei1, dinv1);
    deg_kernel<<<EE / 256, 256, 0, stream>>>(ei2, dinv2);
    dinv_kernel<<<NN / 256, 256, 0, stream>>>(dinv1);
    dinv_kernel<<<NN / 256, 256, 0, stream>>>(dinv2);

    const float* Wl[3] = { W1, W2, W3 };
    const float* bl[3] = { b1, b2, b3 };
    const int din[3]  = { DIN, F1D, F2D };
    const int dou[3]  = { F1D, F2D, F3D };

    const float* feats[2] = { feat1, feat2 };
    const int*   eis[2]   = { ei1, ei2 };
    float*       As[2]    = { A1, A2 };
    float*       dis[2]   = { dinv1, dinv2 };

    for (int g = 0; g < 2; ++g) {
        const float* x = feats[g];
        for (int l = 0; l < 3; ++l) {
            int Di = din[l], Do = dou[l];
            // WMMA GEMM: 8 waves/block = 8 M-tiles; grid (N-tiles, M-tile groups)
            gemm_wmma_kernel<<<dim3(Do / 16, NN / 128), 256, 0, stream>>>(x, Wl[l], Bs,
                                                                          Di, Do);
            int tot = NN * Do;
            initout_kernel<<<tot / 256, 256, 0, stream>>>(Bs, dis[g], bl[l], As[g], Do);
            int etot = EE * (Do >> 2);
            agg_kernel<<<etot / 256, 256, 0, stream>>>(eis[g], eis[g] + EE, Bs, dis[g],
                                                       As[g], Do);
            if (l < 2) relu_kernel<<<tot / 256, 256, 0, stream>>>(As[g], tot);
            x = As[g];
        }
    }

    colmean_kernel<<<64, 256, 0, stream>>>(A1, m1);
    colmean_kernel<<<64, 256, 0, stream>>>(A2, m2);
    tg_kernel<<<1, 32, 0, stream>>>(m1, attW, tg1);
    tg_kernel<<<1, 32, 0, stream>>>(m2, attW, tg2);
    attpool_kernel<<<NN / 256, 256, 0, stream>>>(A1, tg1, p1);
    attpool_kernel<<<NN / 256, 256, 0, stream>>>(A2, tg2, p2);

    // fused WMMA sim passes: pass1 min/max, pass2 histogram (never materialize NxN)
    dim3 sgrid(NN / 256, NN / 32);   // 8 waves/block, each wave = 32x32 output
    sim_tile_kernel<false><<<sgrid, 256, 0, stream>>>(A1, A2, gmin, gmax, hist);
    sim_tile_kernel<true ><<<sgrid, 256, 0, stream>>>(A1, A2, gmin, gmax, hist);

    final_kernel<<<1, 64, 0, stream>>>(p1, p2, tW, tWb, tb, fcW, fcb, scW, scb,
                                       hist, out);
}